// SAGENet_10797547782307
// MI455X (gfx1250) — compile-verified
//
#include <hip/hip_runtime.h>

// ---------------------------------------------------------------------------
// GraphSAGE 3-layer forward for MI455X (gfx1250, wave32, WMMA).
//   h = relu(sage(x)); h = relu(sage(h)); out = sage(h)
//   sage(x) = (segment_mean(x[src], dst)) @ Wl + bl + x @ Wr
// Strategy: edge scatter via f32 atomics (L2-resident agg buffer),
// fused dual-GEMM via V_WMMA_F32_16X16X4_F32, one 16x16 tile per wave.
// ---------------------------------------------------------------------------

#define N_NODES 50000
#define N_EDGES 800000
#define IN_DIM  128
#define HID_DIM 128
#define OUT_DIM 64

typedef __attribute__((ext_vector_type(2))) float v2f;
typedef __attribute__((ext_vector_type(8))) float v8f;

// ---------------------------------------------------------------- utilities
__global__ __launch_bounds__(256) void zero_f32(float* __restrict__ p, int n) {
    int i = blockIdx.x * 256 + threadIdx.x;
    if (i < n) p[i] = 0.0f;
}

__global__ __launch_bounds__(256) void edge_degree(const int* __restrict__ dst,
                                                   float* __restrict__ deg, int nE) {
    int e = blockIdx.x * 256 + threadIdx.x;
    if (e < nE) atomicAdd(&deg[dst[e]], 1.0f);
}

__global__ __launch_bounds__(256) void inv_degree(float* __restrict__ deg, int n) {
    int i = blockIdx.x * 256 + threadIdx.x;
    if (i < n) deg[i] = 1.0f / fmaxf(deg[i], 1.0f);
}

// One wave (32 lanes) per edge; each lane moves 4 contiguous floats (float4
// load, 4x global_atomic_add_f32).  feat is always 128-wide in this net.
__global__ __launch_bounds__(256) void sage_scatter(const float* __restrict__ feat,
                                                    const int* __restrict__ src,
                                                    const int* __restrict__ dst,
                                                    float* __restrict__ agg, int nE) {
    int wave = (blockIdx.x * 256 + threadIdx.x) >> 5;
    if (wave >= nE) return;
    int lane = threadIdx.x & 31;
    int s = src[wave];
    int d = dst[wave];
    const float4* fv = (const float4*)(feat + (size_t)s * 128);
    float4 v = fv[lane];
    float* o = agg + (size_t)d * 128 + lane * 4;
    atomicAdd(o + 0, v.x);
    atomicAdd(o + 1, v.y);
    atomicAdd(o + 2, v.z);
    atomicAdd(o + 3, v.w);
}

// ------------------------------------------------------------ fused WMMA GEMM
// C[row, :] = relu?( (Aagg[row,:]*invdeg[row]) @ Wl + Ax[row,:] @ Wr + bias )
// K = 128 per half.  One wave computes one 16x16 tile of C using
// V_WMMA_F32_16X16X4_F32 (f32 A/B, f32 accum), 64 WMMAs per tile.
//
// Per ISA layouts (wave32):
//   A 16x4 f32 : lane holds row m=lane&15, K-pair (2h, 2h+1), h=lane>>4 -> v2f
//   B 4x16 f32 : lane holds col n=lane&15, same K-pair                 -> v2f
//   C/D 16x16  : VGPR j -> row j + 8h, col = lane&15                   -> v8f
__global__ __launch_bounds__(256) void sage_gemm_wmma(
    const float* __restrict__ Aagg, const float* __restrict__ Ax,
    const float* __restrict__ invdeg,
    const float* __restrict__ Wl, const float* __restrict__ Wr,
    const float* __restrict__ bias, float* __restrict__ C,
    int N, int relu)
{
    const int M_TILES = N_NODES / 16;          // 3125 (exact)
    const int tilesN  = N >> 4;                // 8 or 4
    int tile = blockIdx.x * 8 + (threadIdx.x >> 5);
    if (tile >= M_TILES * tilesN) return;      // wave-uniform exit
    const int tm = tile / tilesN;
    const int tn = tile % tilesN;
    const int lane = threadIdx.x & 31;
    const int mr = lane & 15;                  // A row within tile / B,C col
    const int h  = lane >> 4;                  // K-pair selector
    const int row = tm * 16 + mr;
    const int col = tn * 16 + mr;

    const float2* arow = (const float2*)(Aagg + (size_t)row * 128);
    const float2* xrow = (const float2*)(Ax   + (size_t)row * 128);
    const float   s    = invdeg[row];

    v8f acc = {};

    // ---- half 1: (agg * invdeg) @ Wl, K = 0..127 ----
#pragma unroll
    for (int k = 0; k < 128; k += 4) {
        float2 av = arow[(k >> 1) + h];        // A[row][k+2h], A[row][k+2h+1]
        v2f a; a.x = av.x * s; a.y = av.y * s;
        int kb = k + 2 * h;
        v2f b; b.x = Wl[kb * N + col]; b.y = Wl[(kb + 1) * N + col];
        acc = __builtin_amdgcn_wmma_f32_16x16x4_f32(
            false, a, false, b, (short)0, acc, false, false);
    }
    // ---- half 2: x @ Wr, K = 128..255 ----
#pragma unroll
    for (int k = 0; k < 128; k += 4) {
        float2 av = xrow[(k >> 1) + h];
        v2f a; a.x = av.x; a.y = av.y;
        int kb = k + 2 * h;
        v2f b; b.x = Wr[kb * N + col]; b.y = Wr[(kb + 1) * N + col];
        acc = __builtin_amdgcn_wmma_f32_16x16x4_f32(
            false, a, false, b, (short)0, acc, false, false);
    }

    const float bv = bias[col];
#pragma unroll
    for (int j = 0; j < 8; ++j) {
        float v = acc[j] + bv;
        if (relu) v = fmaxf(v, 0.0f);
        C[(size_t)(tm * 16 + j + 8 * h) * N + col] = v;
    }
}

// ---------------------------------------------------------------------------
extern "C" void kernel_launch(void* const* d_in, const int* in_sizes, int n_in,
                              void* d_out, int out_size, void* d_ws, size_t ws_size,
                              hipStream_t stream) {
    const float* x    = (const float*)d_in[0];
    const int*   edge = (const int*)d_in[1];
    const int*   src  = edge;            // edge_index[0]
    const int*   dst  = edge + N_EDGES;  // edge_index[1]
    const float* Wl0 = (const float*)d_in[2];
    const float* bl0 = (const float*)d_in[3];
    const float* Wr0 = (const float*)d_in[4];
    const float* Wl1 = (const float*)d_in[5];
    const float* bl1 = (const float*)d_in[6];
    const float* Wr1 = (const float*)d_in[7];
    const float* Wl2 = (const float*)d_in[8];
    const float* bl2 = (const float*)d_in[9];
    const float* Wr2 = (const float*)d_in[10];
    float* out = (float*)d_out;

    // workspace layout (floats): invdeg | agg | h1 | h2   (~77 MB)
    float* ws  = (float*)d_ws;
    float* deg = ws;                       // 50048 slot (256B aligned blocks)
    float* agg = deg + 50048;              // 50000*128
    float* h1  = agg + (size_t)N_NODES * 128;
    float* h2  = h1  + (size_t)N_NODES * 128;

    const int ZB_N   = (N_NODES + 255) / 256;
    const int ZB_AGG = (N_NODES * 128 + 255) / 256;
    const int EB     = (N_EDGES + 255) / 256;          // 1 thread / edge
    const int SB     = (N_EDGES * 32 + 255) / 256;     // 1 wave / edge
    const int GB128  = ((N_NODES / 16) * (HID_DIM / 16) + 7) / 8;
    const int GB64   = ((N_NODES / 16) * (OUT_DIM / 16) + 7) / 8;

    // degree -> 1/max(deg,1), computed once (same edges every layer)
    zero_f32<<<ZB_N, 256, 0, stream>>>(deg, N_NODES);
    edge_degree<<<EB, 256, 0, stream>>>(dst, deg, N_EDGES);
    inv_degree<<<ZB_N, 256, 0, stream>>>(deg, N_NODES);

    // ---- layer 0: x -> h1 (relu) ----
    zero_f32<<<ZB_AGG, 256, 0, stream>>>(agg, N_NODES * 128);
    sage_scatter<<<SB, 256, 0, stream>>>(x, src, dst, agg, N_EDGES);
    sage_gemm_wmma<<<GB128, 256, 0, stream>>>(agg, x, deg, Wl0, Wr0, bl0, h1,
                                              HID_DIM, 1);

    // ---- layer 1: h1 -> h2 (relu) ----
    zero_f32<<<ZB_AGG, 256, 0, stream>>>(agg, N_NODES * 128);
    sage_scatter<<<SB, 256, 0, stream>>>(h1, src, dst, agg, N_EDGES);
    sage_gemm_wmma<<<GB128, 256, 0, stream>>>(agg, h1, deg, Wl1, Wr1, bl1, h2,
                                              HID_DIM, 1);

    // ---- layer 2: h2 -> out (no relu) ----
    zero_f32<<<ZB_AGG, 256, 0, stream>>>(agg, N_NODES * 128);
    sage_scatter<<<SB, 256, 0, stream>>>(h2, src, dst, agg, N_EDGES);
    sage_gemm_wmma<<<GB64, 256, 0, stream>>>(agg, h2, deg, Wl2, Wr2, bl2, out,
                                             OUT_DIM, 0);
}